// SingleQubit_FuzzyLayer_89816356094199
// MI455X (gfx1250) — compile-verified
//
#include <hip/hip_runtime.h>
#include <math.h>

// ---------------------------------------------------------------------------
// SingleQubit fuzzy layer, collapsed algebraically:
//   u0(x) = c^3 a0 + c^2 s a1 + c s^2 a2 + s^3 a3   (c=cos(x/2), s=sin(x/2))
//   qout  = Re(u0)^2 + Im(u0)^2
// a_k are per-class complex constants derived from params only.
// Per-pixel contraction [1x4]x[4x20] is batched 16 pixels at a time through
// V_WMMA_F32_16X16X4_F32 (A = coeff rows / outputs, B = monomials / pixels).
// Memory-bound: ~6.3MB read + ~62.9MB write -> ~3us floor at 23.3 TB/s.
// ---------------------------------------------------------------------------

typedef __attribute__((ext_vector_type(2))) float v2f;
typedef __attribute__((ext_vector_type(8))) float v8f;

#define NUM_CLASSES 10
#define N_LAYERS 3

// ws layout: Acoef[2][16][4]  (tile, row = output index, K = monomial index)
// Output index n: n=2c -> Re coeffs of class c, n=2c+1 -> Im coeffs.
// Tile 0 = outputs 0..15 (classes 0..7), tile 1 rows 0..3 = outputs 16..19
// (classes 8..9), remaining rows zero.
__global__ void qc_setup_coeffs(const float* __restrict__ params,
                                float* __restrict__ Acoef) {
  int t = threadIdx.x;
  if (t < 2 * 16 * 4) Acoef[t] = 0.0f;   // zero both tiles first
  __syncthreads();
  if (t < NUM_CLASSES) {
    // State polynomial in (c,s): w[j] <-> monomial c^(deg-j) s^j,
    // each w[j] is a complex 2-vector (top = tr+i*ti, bottom = br+i*bi).
    float tr[4] = {1.f, 0.f, 0.f, 0.f};
    float ti[4] = {0.f, 0.f, 0.f, 0.f};
    float br[4] = {0.f, 0.f, 0.f, 0.f};
    float bi[4] = {0.f, 0.f, 0.f, 0.f};
    int len = 1;
    for (int l = 0; l < N_LAYERS; ++l) {
      // Multiply by Ry(x) = c*I + s*J,  J = [[0,-1],[1,0]]:
      // new_w[j] = w[j] + J*w[j-1]
      float ntr[4], nti[4], nbr[4], nbi[4];
      for (int j = 0; j <= len; ++j) {
        float xr = 0.f, xi = 0.f, yr = 0.f, yi = 0.f;
        if (j < len) { xr = tr[j]; xi = ti[j]; yr = br[j]; yi = bi[j]; }
        if (j >= 1)  { xr -= br[j-1]; xi -= bi[j-1]; yr += tr[j-1]; yi += ti[j-1]; }
        ntr[j] = xr; nti[j] = xi; nbr[j] = yr; nbi[j] = yi;
      }
      ++len;
      // Constant gate G_l = Rz(p2) * Ry(p1) * Rz(p0), half-angles th=0.5*p.
      const float* P = params + (t * N_LAYERS + l) * 3;
      float th0 = 0.5f * P[0], th1 = 0.5f * P[1], th2 = 0.5f * P[2];
      float s1, c1, sz0, cz0, sz2, cz2;
      sincosf(th1, &s1, &c1);
      sincosf(th0, &sz0, &cz0);
      sincosf(th2, &sz2, &cz2);
      for (int j = 0; j < len; ++j) {
        float xr = ntr[j], xi = nti[j], yr = nbr[j], yi = nbi[j];
        // Rz(th0): top *= e^{-i th0}, bottom *= e^{+i th0}
        float txr = xr * cz0 + xi * sz0;
        float txi = xi * cz0 - xr * sz0;
        float tyr = yr * cz0 - yi * sz0;
        float tyi = yi * cz0 + yr * sz0;
        // Ry(th1)
        float uxr = c1 * txr - s1 * tyr, uxi = c1 * txi - s1 * tyi;
        float uyr = s1 * txr + c1 * tyr, uyi = s1 * txi + c1 * tyi;
        // Rz(th2)
        tr[j] = uxr * cz2 + uxi * sz2;
        ti[j] = uxi * cz2 - uxr * sz2;
        br[j] = uyr * cz2 - uyi * sz2;
        bi[j] = uyi * cz2 + uyr * sz2;
      }
    }
    // a_k = top component of w[k]; write Re row (n=2t) and Im row (n=2t+1)
    int nRe = 2 * t, nIm = 2 * t + 1;
    for (int k = 0; k < 4; ++k) {
      Acoef[((nRe >> 4) * 16 + (nRe & 15)) * 4 + k] = tr[k];
      Acoef[((nIm >> 4) * 16 + (nIm & 15)) * 4 + k] = ti[k];
    }
  }
}

// Main kernel: each wave handles tiles of 16 pixels.
//  A frag (constant): lane L -> row M=L%16; V0=A[M][K0], V1=A[M][K0+1],
//                     K0 = 0 for lanes 0..15, 2 for lanes 16..31.
//  B frag: lane L -> column N=L%16 (pixel); V0/V1 = monomials {c^3,c^2 s}
//          (lanes 0..15) or {c s^2, s^3} (lanes 16..31).
//  D: lane L<16 vgpr v = u[v] of pixel L; lane L>=16 vgpr v = u[v+8].
//  => all Re/Im pairs land in one lane; squares summed in-register.
__global__ void __launch_bounds__(256)
qc_fuzzy_kernel(const float* __restrict__ X, const float* __restrict__ Acoef,
                float* __restrict__ out, int ntiles) {
  const int lane = threadIdx.x & 31;
  const int wave = threadIdx.x >> 5;
  const int wpb  = blockDim.x >> 5;
  const int wid  = blockIdx.x * wpb + wave;
  const int wstride = gridDim.x * wpb;

  const int  n  = lane & 15;
  const bool hi = lane >= 16;
  const int  kb = hi ? 2 : 0;

  v2f a0, a1;  // constant coefficient fragments (tile 0, tile 1)
  a0[0] = Acoef[(0 * 16 + n) * 4 + kb];
  a0[1] = Acoef[(0 * 16 + n) * 4 + kb + 1];
  a1[0] = Acoef[(1 * 16 + n) * 4 + kb];
  a1[1] = Acoef[(1 * 16 + n) * 4 + kb + 1];

  for (int t = wid; t < ntiles; t += wstride) {
    const int p = t * 16 + n;            // pixel for this lane (dup per half)
    const int tn = t + wstride;
    if (tn < ntiles) __builtin_prefetch(X + tn * 16 + n, 0, 0);

    float x = X[p];
    float s, c;
    __sincosf(0.5f * x, &s, &c);         // args in [0, 0.5): fast path is exact enough
    float c2 = c * c, s2 = s * s;
    v2f b;
    b[0] = hi ? c * s2 : c2 * c;         // K=2 : K=0
    b[1] = hi ? s2 * s : c2 * s;         // K=3 : K=1

    v8f z = {};
    v8f d0 = __builtin_amdgcn_wmma_f32_16x16x4_f32(false, a0, false, b,
                                                   (short)0, z, false, false);
    v8f d1 = __builtin_amdgcn_wmma_f32_16x16x4_f32(false, a1, false, b,
                                                   (short)0, z, false, false);

    float* o = out + (size_t)p * NUM_CLASSES;   // p*10 even -> 8B aligned
    if (!hi) {
      // outputs 0..7 (classes 0..3) + outputs 16..19 (classes 8..9)
      float q0 = d0[0] * d0[0] + d0[1] * d0[1];
      float q1 = d0[2] * d0[2] + d0[3] * d0[3];
      float q2 = d0[4] * d0[4] + d0[5] * d0[5];
      float q3 = d0[6] * d0[6] + d0[7] * d0[7];
      float q8 = d1[0] * d1[0] + d1[1] * d1[1];
      float q9 = d1[2] * d1[2] + d1[3] * d1[3];
      *reinterpret_cast<float2*>(o + 0) = make_float2(q0, q1);
      *reinterpret_cast<float2*>(o + 2) = make_float2(q2, q3);
      *reinterpret_cast<float2*>(o + 8) = make_float2(q8, q9);
    } else {
      // outputs 8..15 (classes 4..7)
      float q4 = d0[0] * d0[0] + d0[1] * d0[1];
      float q5 = d0[2] * d0[2] + d0[3] * d0[3];
      float q6 = d0[4] * d0[4] + d0[5] * d0[5];
      float q7 = d0[6] * d0[6] + d0[7] * d0[7];
      *reinterpret_cast<float2*>(o + 4) = make_float2(q4, q5);
      *reinterpret_cast<float2*>(o + 6) = make_float2(q6, q7);
    }
  }
}

extern "C" void kernel_launch(void* const* d_in, const int* in_sizes, int n_in,
                              void* d_out, int out_size, void* d_ws, size_t ws_size,
                              hipStream_t stream) {
  const float* x      = (const float*)d_in[0];   // [512,3,32,32] f32
  const float* params = (const float*)d_in[1];   // [10,3,3] f32
  float* out   = (float*)d_out;                  // [512,3072,10] f32
  float* Acoef = (float*)d_ws;                   // 2*16*4 floats

  const int npix   = in_sizes[0];                // 1,572,864 (multiple of 16)
  const int ntiles = npix / 16;

  hipLaunchKernelGGL(qc_setup_coeffs, dim3(1), dim3(128), 0, stream,
                     params, Acoef);

  int blocks = (ntiles + 7) / 8;                 // 8 waves per 256-thread block
  if (blocks > 4096) blocks = 4096;
  hipLaunchKernelGGL(qc_fuzzy_kernel, dim3(blocks), dim3(256), 0, stream,
                     x, Acoef, out, ntiles);
}